// TensorConNet_28948079575659
// MI455X (gfx1250) — compile-verified
//
#include <hip/hip_runtime.h>
#include <hip/hip_bf16.h>

// ---- problem constants (from reference) ----
#define NVIEW 9
#define CCH   24      // image feature channels C
#define IMH   120
#define IMW   160
#define NX    96      // full voxel grid edge
#define ND    24      // downsampled edge (96/4)
#define SP    96      // conv spatial edge
#define SP2   9216    // 96*96
#define OC    192     // conv output channels (CH2D * ND)
#define CIN1  576     // conv1 input channels (C * ND)
#define CIN2  192     // conv2 input channels

typedef __attribute__((ext_vector_type(2))) float v2f;
typedef __attribute__((ext_vector_type(8))) float v8f;

// Channel-pair-interleaved activation layout used everywhere:
//   buf[((ch>>1)*SP2 + pos)*2 + (ch&1)]     (pos = a*SP + b)
// so the two K-values a WMMA lane needs are one contiguous float2 (b64 load).

__device__ __forceinline__ v8f wmma4(v2f a, v2f b, v8f c) {
    return __builtin_amdgcn_wmma_f32_16x16x4_f32(false, a, false, b,
                                                 (short)0, c, false, false);
}

// ---------------------------------------------------------------------------
// Pack W[oc][ic][ky][kx] -> pair-interleaved K-major:
//   Apack[(((kk*Cin+ic)>>1)*OC + oc)*2 + (ic&1)],  kk = ky*3+kx
// A-fragment = one b64 load, coalesced across the 16 oc lanes.
// ---------------------------------------------------------------------------
__global__ void pack_weights_kernel(const float* __restrict__ Wsrc,
                                    float* __restrict__ Apack, int Cin) {
    int t = blockIdx.x * blockDim.x + threadIdx.x;
    int total = 9 * Cin * OC;
    if (t >= total) return;
    int oc  = t % OC;
    int rem = t / OC;
    int ic  = rem % Cin;
    int kk  = rem / Cin;
    int ky = kk / 3, kx = kk - ky * 3;
    int k  = kk * Cin + ic;
    Apack[(size_t)((k >> 1) * OC + oc) * 2 + (ic & 1)] =
        Wsrc[(((size_t)oc * Cin + ic) * 3 + ky) * 3 + kx];
}

// ---------------------------------------------------------------------------
// Back-projection: one thread per downsampled voxel; writes conv input in the
// pair-interleaved layout (fusing the reference transpose/reshape) + counts.
// ---------------------------------------------------------------------------
__global__ void backproj_kernel(const float* __restrict__ feats,
                                const float* __restrict__ proj,
                                const float* __restrict__ origin,
                                float* __restrict__ Xout,
                                float* __restrict__ cntout,
                                int axis) {
    int t = blockIdx.x * blockDim.x + threadIdx.x;
    int n0 = (axis == 0) ? ND : SP;
    int n1 = (axis == 1) ? ND : SP;
    int n2 = (axis == 2) ? ND : SP;
    if (t >= n0 * n1 * n2) return;
    int i2 = t % n2;
    int i1 = (t / n2) % n1;
    int i0 = t / (n1 * n2);

    float vs0 = (axis == 0) ? 0.16f : 0.04f;
    float vs1 = (axis == 1) ? 0.16f : 0.04f;
    float vs2 = (axis == 2) ? 0.16f : 0.04f;
    float wx = origin[0] + i0 * vs0;
    float wy = origin[1] + i1 * vs1;
    float wz = origin[2] + i2 * vs2;

    float acc[CCH];
#pragma unroll
    for (int c = 0; c < CCH; c++) acc[c] = 0.f;
    int cnt = 0;

    for (int v = 0; v < NVIEW; v++) {
        const float* P = proj + v * 16;
        float cx = P[0] * wx + P[1] * wy + P[2]  * wz + P[3];
        float cy = P[4] * wx + P[5] * wy + P[6]  * wz + P[7];
        float cz = P[8] * wx + P[9] * wy + P[10] * wz + P[11];
        float px = cx / cz, py = cy / cz;
        bool ok = (px >= 0.f) && (px <= (float)(IMW - 1)) &&
                  (py >= 0.f) && (py <= (float)(IMH - 1)) && (cz > 0.f);
        if (!ok) continue;
        cnt++;
        float fx0 = floorf(px), fy0 = floorf(py);
        float bx = px - fx0, by = py - fy0;
        int x0 = min(max((int)fx0, 0), IMW - 1);
        int x1 = min(x0 + 1, IMW - 1);
        int y0 = min(max((int)fy0, 0), IMH - 1);
        int y1 = min(y0 + 1, IMH - 1);
        float w00 = (1.f - bx) * (1.f - by);
        float w01 = bx * (1.f - by);
        float w10 = (1.f - bx) * by;
        float w11 = bx * by;
        const float* fv = feats + (size_t)v * CCH * IMH * IMW;
#pragma unroll
        for (int c = 0; c < CCH; c++) {
            const float* fc = fv + (size_t)c * IMH * IMW;
            acc[c] += fc[y0 * IMW + x0] * w00 + fc[y0 * IMW + x1] * w01 +
                      fc[y1 * IMW + x0] * w10 + fc[y1 * IMW + x1] * w11;
        }
    }

    int d, a, b;
    if (axis == 0)      { d = i0; a = i1; b = i2; }
    else if (axis == 1) { d = i1; a = i0; b = i2; }
    else                { d = i2; a = i0; b = i1; }

    int pos = a * SP + b;
    float inv = 1.f / (float)max(cnt, 1);
#pragma unroll
    for (int c = 0; c < CCH; c++) {
        int ch = c * ND + d;
        Xout[(size_t)((ch >> 1) * SP2 + pos) * 2 + (ch & 1)] = acc[c] * inv;
    }
    cntout[d * SP2 + pos] = (float)cnt;
}

// ---------------------------------------------------------------------------
// 3x3 SAME conv as implicit GEMM on V_WMMA_F32_16X16X4_F32.
// One wave computes a 32(oc) x 32(spatial) tile = 2x2 accumulator tiles.
// Per K-step (K+=4): 2 A b64 loads + 2 B b64 loads feed 4 WMMAs.
// A-frag: lane l -> A[M=l%16][k0+2*(l/16)+{0,1}]; B mirrors the K split.
// Padding (rows AND columns) folded into per-lane 0/1 operand masks so the
// tap loop is branch-free (EXEC all-ones at every WMMA).
// CIN/RELU are template params so the ic-loop trip count (144 / 48) is
// static: no remainder path, no accumulator register-set merges.
// ---------------------------------------------------------------------------
template <int CIN, int RELU>
__global__ void conv3x3_wmma_kernel(const float* __restrict__ X,
                                    const float* __restrict__ Apack,
                                    const float* __restrict__ bias,
                                    float* __restrict__ Y) {
    int lane = threadIdx.x & 31;
    int wave = blockIdx.x * (blockDim.x >> 5) + (threadIdx.x >> 5);
    const int MT = OC / 32;                  // 6 oc super-tiles
    int mt = wave % MT;
    int nt = wave / MT;
    if (nt >= SP2 / 32) return;

    int oc_base = mt * 32;
    int p_base  = nt * 32;
    int a  = p_base / SP;                    // 96 % 32 == 0: row never crossed
    int b0 = p_base % SP;
    int m = lane & 15;
    int h = lane >> 4;

    v8f acc00 = {0.f,0.f,0.f,0.f,0.f,0.f,0.f,0.f};
    v8f acc01 = acc00, acc10 = acc00, acc11 = acc00;

    const v2f* Xp = (const v2f*)X;
    const v2f* Ap = (const v2f*)Apack;

    for (int kk = 0; kk < 9; kk++) {
        int ky = kk / 3, kx = kk - ky * 3;
        int ay = a + ky - 1;
        float rmask = (ay >= 0 && ay < SP) ? 1.f : 0.f;   // wave-uniform
        int ayc = min(max(ay, 0), SP - 1);
        int bx0 = b0 + m + kx - 1;
        int bx1 = bx0 + 16;
        float m0 = ((bx0 >= 0 && bx0 < SP) ? rmask : 0.f);
        float m1 = ((bx1 < SP) ? rmask : 0.f);            // bx1 >= 15 always
        int bc0 = min(max(bx0, 0), SP - 1);
        int bc1 = min(bx1, SP - 1);
        const v2f* Xr0 = Xp + (size_t)(ayc * SP + bc0);
        const v2f* Xr1 = Xp + (size_t)(ayc * SP + bc1);
        const v2f* A0  = Ap + (size_t)(kk * (CIN >> 1)) * OC + oc_base + m;

#pragma unroll 4
        for (int ic = 0; ic < CIN; ic += 4) {
            int pp = (ic >> 1) + h;          // channel-pair index
            v2f a0 = A0[(size_t)pp * OC];
            v2f a1 = A0[(size_t)pp * OC + 16];
            v2f bv0 = Xr0[(size_t)pp * SP2];
            v2f bv1 = Xr1[(size_t)pp * SP2];
            bv0.x *= m0; bv0.y *= m0;
            bv1.x *= m1; bv1.y *= m1;
            acc00 = wmma4(a0, bv0, acc00);
            acc01 = wmma4(a0, bv1, acc01);
            acc10 = wmma4(a1, bv0, acc10);
            acc11 = wmma4(a1, bv1, acc11);
        }
    }

    // Store: acc[r] of lane -> C[M = r + 8h][N = m]; rows (r, r+1) share an
    // oc-pair -> contiguous float2 store in the pair-interleaved layout.
    v2f* Yp = (v2f*)Y;
    int col0 = b0 + m, col1 = col0 + 16;
#pragma unroll
    for (int oi = 0; oi < 2; oi++) {
        v8f va = oi ? acc10 : acc00;
        v8f vb = oi ? acc11 : acc01;
        int ob = oc_base + oi * 16 + 8 * h;  // even
#pragma unroll
        for (int r = 0; r < 8; r += 2) {
            int oc = ob + r;
            float e0a = va[r]   + bias[oc];
            float e1a = va[r+1] + bias[oc + 1];
            float e0b = vb[r]   + bias[oc];
            float e1b = vb[r+1] + bias[oc + 1];
            if (RELU) {
                e0a = fmaxf(e0a, 0.f); e1a = fmaxf(e1a, 0.f);
                e0b = fmaxf(e0b, 0.f); e1b = fmaxf(e1b, 0.f);
            }
            v2f sa = {e0a, e1a}, sb = {e0b, e1b};
            Yp[(size_t)(oc >> 1) * SP2 + a * SP + col0] = sa;
            Yp[(size_t)(oc >> 1) * SP2 + a * SP + col1] = sb;
        }
    }
}

// ---------------------------------------------------------------------------
// Head: fuse DOWN-repeat + channel concat + tsdf/occ linear heads + mask.
// Reads the pair-interleaved conv2 outputs.
// ---------------------------------------------------------------------------
__global__ void head_kernel(const float* __restrict__ Y0,
                            const float* __restrict__ Y1,
                            const float* __restrict__ Y2,
                            const float* __restrict__ C0,
                            const float* __restrict__ C1,
                            const float* __restrict__ C2,
                            const float* __restrict__ tw, const float* __restrict__ tb,
                            const float* __restrict__ ow, const float* __restrict__ ob,
                            float* __restrict__ out) {
    int n = blockIdx.x * blockDim.x + threadIdx.x;
    const int NTOT = NX * NX * NX;
    if (n >= NTOT) return;
    int z = n % NX;
    int y = (n / NX) % NX;
    int x = n / (NX * NX);

    float f[24];
#pragma unroll
    for (int c = 0; c < 8; c++) {
        int ch = c * ND + (x >> 2);
        f[c] = Y0[(size_t)((ch >> 1) * SP2 + y * SP + z) * 2 + (ch & 1)];
    }
#pragma unroll
    for (int c = 0; c < 8; c++) {
        int ch = c * ND + (y >> 2);
        f[8 + c] = Y1[(size_t)((ch >> 1) * SP2 + x * SP + z) * 2 + (ch & 1)];
    }
#pragma unroll
    for (int c = 0; c < 8; c++) {
        int ch = c * ND + (z >> 2);
        f[16 + c] = Y2[(size_t)((ch >> 1) * SP2 + x * SP + y) * 2 + (ch & 1)];
    }

    float ts = tb[0], oc = ob[0];
#pragma unroll
    for (int c = 0; c < 24; c++) { ts += f[c] * tw[c]; oc += f[c] * ow[c]; }
    out[(size_t)n * 2 + 0] = ts;
    out[(size_t)n * 2 + 1] = oc;

    float c0 = C0[(x >> 2) * SP2 + y * SP + z];
    float c1 = C1[(y >> 2) * SP2 + x * SP + z];
    float c2 = C2[(z >> 2) * SP2 + x * SP + y];
    out[(size_t)NTOT * 2 + n] = (c0 > 1.f || c1 > 1.f || c2 > 1.f) ? 1.f : 0.f;
}

// ---------------------------------------------------------------------------
extern "C" void kernel_launch(void* const* d_in, const int* in_sizes, int n_in,
                              void* d_out, int out_size, void* d_ws, size_t ws_size,
                              hipStream_t stream) {
    const float* feats  = (const float*)d_in[0];
    const float* proj   = (const float*)d_in[1];
    const float* origin = (const float*)d_in[2];
    const float* W1 = (const float*)d_in[3];
    const float* b1 = (const float*)d_in[4];
    const float* W2 = (const float*)d_in[5];
    const float* b2 = (const float*)d_in[6];
    const float* tw = (const float*)d_in[7];
    const float* tb = (const float*)d_in[8];
    const float* ow = (const float*)d_in[9];
    const float* ob = (const float*)d_in[10];
    float* out = (float*)d_out;

    // workspace layout (floats); conv2 output aliases the conv1 input buffer
    float* ws = (float*)d_ws;
    size_t off = 0;
    float* A1 = ws + off; off += (size_t)9 * CIN1 * OC;   //  3.98 MB
    float* A2 = ws + off; off += (size_t)9 * CIN2 * OC;   //  1.33 MB
    float* Xb[3], *Y1b[3], *Cb[3];
    for (int i = 0; i < 3; i++) { Xb[i]  = ws + off; off += (size_t)CIN1 * SP2; }
    for (int i = 0; i < 3; i++) { Y1b[i] = ws + off; off += (size_t)OC * SP2;   }
    for (int i = 0; i < 3; i++) { Cb[i]  = ws + off; off += (size_t)ND * SP2;   }
    // total ~88.6 MB

    pack_weights_kernel<<<(9 * CIN1 * OC + 255) / 256, 256, 0, stream>>>(W1, A1, CIN1);
    pack_weights_kernel<<<(9 * CIN2 * OC + 255) / 256, 256, 0, stream>>>(W2, A2, CIN2);

    for (int ax = 0; ax < 3; ax++)
        backproj_kernel<<<(ND * SP2 + 255) / 256, 256, 0, stream>>>(
            feats, proj, origin, Xb[ax], Cb[ax], ax);

    const int waves = (OC / 32) * (SP2 / 32);    // 1728
    const int cblks = waves / 8;                 // 216 blocks of 8 waves
    for (int ax = 0; ax < 3; ax++) {
        conv3x3_wmma_kernel<CIN1, 1><<<cblks, 256, 0, stream>>>(Xb[ax], A1, b1, Y1b[ax]);
        conv3x3_wmma_kernel<CIN2, 0><<<cblks, 256, 0, stream>>>(Y1b[ax], A2, b2, Xb[ax]);
    }

    head_kernel<<<(NX * NX * NX + 255) / 256, 256, 0, stream>>>(
        Xb[0], Xb[1], Xb[2], Cb[0], Cb[1], Cb[2], tw, tb, ow, ob, out);
}